// LearnableSplines_36086315221619
// MI455X (gfx1250) — compile-verified
//
#include <hip/hip_runtime.h>
#include <stdint.h>
#include <stddef.h>

typedef __attribute__((ext_vector_type(4))) float v4f;

#define D_DIM 1024   // embed dim (kernels specialized for this)
#define TPB   256    // threads per block = 8 wave32; each thread owns 4 columns
#define SPB   8      // samples per block

// ---------------------------------------------------------------------------
// CDNA5 async global->LDS copy (ASYNCcnt-tracked DMA path, gfx1250).
// LDS destination address = low 32 bits of the flat shared-aperture pointer
// (ISA 10.2).  ptrtoint escapes the shared array
// and the "memory" clobber orders it against normal ds loads.
// ---------------------------------------------------------------------------
__device__ __forceinline__ void async_copy_b128(const float* gsrc, const float* lds_dst) {
  uint32_t lds_off = (uint32_t)(uintptr_t)lds_dst;
  asm volatile("global_load_async_to_lds_b128 %0, %1, off"
               :
               : "v"(lds_off), "v"((uint64_t)(uintptr_t)gsrc)
               : "memory");
}
#define WAIT_ASYNCCNT(n) asm volatile("s_wait_asynccnt " #n ::: "memory")

// ---------------------------------------------------------------------------
// Per-sample spline setup: segment index, gathered control rows, weights.
// All addresses are block-uniform -> compiler scalarizes to SMEM/SALU.
// ---------------------------------------------------------------------------
struct SampleW { int i0, i1, i2, i3; float w0, w1, w2, w3; };

__device__ __forceinline__ SampleW setup_sample(float t,
                                                const float* __restrict__ tp,
                                                const float* __restrict__ cc,
                                                int N) {
  SampleW r;
  float scaled = t * (float)(N - 1);
  int seg = (int)floorf(scaled);
  seg = seg < 0 ? 0 : seg;
  seg = seg > N - 2 ? N - 2 : seg;
  float lt = scaled - (float)seg;

  float tens = 1.0f / (1.0f + expf(-tp[seg]));
  float c1 = cc[seg];
  float c2 = cc[seg + 1];

  float t2 = lt * lt;
  float t3 = t2 * lt;
  float v0 = (-0.5f * t3 + t2 - 0.5f * lt) * tens;
  float v1 = ( 1.5f * t3 - 2.5f * t2 + 1.0f) * c1;
  float v2 = (-1.5f * t3 + 2.0f * t2 + 0.5f * lt) * c2;
  float v3 = ( 0.5f * t3 - 0.5f * t2) * tens;
  float total = v0 + v1 + v2 + v3;
  r.w0 = v0 / total; r.w1 = v1 / total; r.w2 = v2 / total; r.w3 = v3 / total;

  r.i0 = (seg - 1 < 0) ? 0 : seg - 1;
  r.i1 = seg;
  r.i2 = seg + 1;
  r.i3 = (seg + 2 > N - 1) ? N - 1 : seg + 2;

  // boundary handling: t<=0 -> emb[0], t>=1 -> emb[N-1]
  if (t <= 0.0f) { r.i0 = r.i1 = r.i2 = r.i3 = 0;     r.w0 = 0.f; r.w1 = 1.f; r.w2 = 0.f; r.w3 = 0.f; }
  if (t >= 1.0f) { r.i0 = r.i1 = r.i2 = r.i3 = N - 1; r.w0 = 0.f; r.w1 = 1.f; r.w2 = 0.f; r.w3 = 0.f; }
  return r;
}

// ---------------------------------------------------------------------------
// Pass 1: emb = (word_embeddings + word_biases) * semantic_weights  (4 MB,
// stays L2-resident: 192 MB L2, each row re-read ~256x by pass 2).
// ---------------------------------------------------------------------------
__global__ void __launch_bounds__(TPB)
prep_emb(const float* __restrict__ we, const float* __restrict__ wb,
         const float* __restrict__ sw, float* __restrict__ emb, int total) {
  int idx = (blockIdx.x * TPB + threadIdx.x) * 4;
  if (idx >= total) return;
  int col = idx & (D_DIM - 1);
  v4f a = *(const v4f*)(we + idx);
  v4f b = *(const v4f*)(wb + idx);
  v4f s = *(const v4f*)(sw + col);
  *(v4f*)(emb + idx) = (a + b) * s;
}

// ---------------------------------------------------------------------------
// Pass 2: per-sample gather + blend.  Double-buffered async global->LDS
// pipeline: stage sample i+1's four 4 KB control rows while blending sample i.
// Each thread reads back exactly the LDS bytes it copied -> no barriers,
// only s_wait_asynccnt.  Output streamed with non-temporal 128-bit stores.
// ---------------------------------------------------------------------------
__device__ __forceinline__ void stage_prefetch(const float* __restrict__ emb,
                                               const SampleW& w,
                                               float (*buf)[D_DIM], int j) {
  async_copy_b128(emb + (size_t)w.i0 * D_DIM + j, &buf[0][j]);
  async_copy_b128(emb + (size_t)w.i1 * D_DIM + j, &buf[1][j]);
  async_copy_b128(emb + (size_t)w.i2 * D_DIM + j, &buf[2][j]);
  async_copy_b128(emb + (size_t)w.i3 * D_DIM + j, &buf[3][j]);
}

__global__ void __launch_bounds__(TPB)
spline_main(const float* __restrict__ emb, const float* __restrict__ tq,
            const float* __restrict__ tp, const float* __restrict__ cc,
            float* __restrict__ out, int N, int S) {
  __shared__ float lds[2][4][D_DIM];   // 32 KB: 2 pipeline stages x 4 rows
  const int j  = threadIdx.x * 4;
  const int s0 = blockIdx.x * SPB;

  {
    SampleW w = setup_sample(tq[s0], tp, cc, N);
    stage_prefetch(emb, w, lds[0], j);
  }

#pragma unroll
  for (int i = 0; i < SPB; ++i) {
    const int buf = i & 1;
    if (i + 1 < SPB) {             // compile-time after unroll
      if (s0 + i + 1 < S) {
        SampleW wn = setup_sample(tq[s0 + i + 1], tp, cc, N);
        stage_prefetch(emb, wn, lds[buf ^ 1], j);
        WAIT_ASYNCCNT(4);          // 4 newer copies may remain in flight
      } else {
        WAIT_ASYNCCNT(0);
      }
    } else {
      WAIT_ASYNCCNT(0);
    }
    if (s0 + i < S) {
      SampleW w = setup_sample(tq[s0 + i], tp, cc, N);
      v4f p0 = *(const v4f*)&lds[buf][0][j];
      v4f p1 = *(const v4f*)&lds[buf][1][j];
      v4f p2 = *(const v4f*)&lds[buf][2][j];
      v4f p3 = *(const v4f*)&lds[buf][3][j];
      v4f acc = p0 * w.w0 + p1 * w.w1 + p2 * w.w2 + p3 * w.w3;
      __builtin_nontemporal_store(acc, (v4f*)(out + (size_t)(s0 + i) * D_DIM + j));
    }
  }
}

// ---------------------------------------------------------------------------
// Fallback (workspace too small): fused single pass, direct global loads.
// ---------------------------------------------------------------------------
__global__ void __launch_bounds__(TPB)
spline_fused(const float* __restrict__ we, const float* __restrict__ wb,
             const float* __restrict__ sw, const float* __restrict__ tq,
             const float* __restrict__ tp, const float* __restrict__ cc,
             float* __restrict__ out, int N, int S) {
  const int j  = threadIdx.x * 4;
  const v4f swv = *(const v4f*)(sw + j);
  const int s0 = blockIdx.x * SPB;
  for (int i = 0; i < SPB; ++i) {
    int s = s0 + i;
    if (s >= S) return;
    SampleW w = setup_sample(tq[s], tp, cc, N);
    const int   idx[4] = {w.i0, w.i1, w.i2, w.i3};
    const float ww[4]  = {w.w0, w.w1, w.w2, w.w3};
    v4f acc = {0.f, 0.f, 0.f, 0.f};
#pragma unroll
    for (int r = 0; r < 4; ++r) {
      v4f a = *(const v4f*)(we + (size_t)idx[r] * D_DIM + j);
      v4f b = *(const v4f*)(wb + (size_t)idx[r] * D_DIM + j);
      acc += (a + b) * swv * ww[r];
    }
    __builtin_nontemporal_store(acc, (v4f*)(out + (size_t)s * D_DIM + j));
  }
}

// ---------------------------------------------------------------------------
extern "C" void kernel_launch(void* const* d_in, const int* in_sizes, int n_in,
                              void* d_out, int out_size, void* d_ws, size_t ws_size,
                              hipStream_t stream) {
  const float* we = (const float*)d_in[0];  // word_embeddings [N, D]
  const float* tq = (const float*)d_in[1];  // t_query        [S]
  const float* tp = (const float*)d_in[2];  // tension_params [N-1]
  const float* sw = (const float*)d_in[3];  // semantic_weights [D]
  const float* wb = (const float*)d_in[4];  // word_biases    [N, D]
  const float* cc = (const float*)d_in[5];  // curvature_controls [N]
  float* out = (float*)d_out;

  const int S = in_sizes[1];
  const int D = in_sizes[3];
  const int N = in_sizes[5];
  const int grid_main = (S + SPB - 1) / SPB;
  const size_t emb_bytes = (size_t)N * (size_t)D * sizeof(float);

  if (D == D_DIM && ws_size >= emb_bytes) {
    float* emb = (float*)d_ws;
    const int total = N * D;
    prep_emb<<<(total / 4 + TPB - 1) / TPB, TPB, 0, stream>>>(we, wb, sw, emb, total);
    spline_main<<<grid_main, TPB, 0, stream>>>(emb, tq, tp, cc, out, N, S);
  } else if (D == D_DIM) {
    spline_fused<<<grid_main, TPB, 0, stream>>>(we, wb, sw, tq, tp, cc, out, N, S);
  }
}